// ScaledDotProductAtten_25769803881
// MI455X (gfx1250) — compile-verified
//
#include <hip/hip_runtime.h>

// Attention: context = softmax(Q K^T) V, plus full attention matrix output.
// B=16, L=2048, D=64, fp32. Uses V_WMMA_F32_16X16X4_F32 (CDNA5 wave32 WMMA).

#define NBATCH 16
#define LSEQ   2048
#define DDIM   64
#define BQ     16           // q-tile rows per workgroup

typedef float v2f __attribute__((ext_vector_type(2)));
typedef float v8f __attribute__((ext_vector_type(8)));

extern "C" __global__ __launch_bounds__(128)
void sdpa_wmma_f32_kernel(const float* __restrict__ Q,
                          const float* __restrict__ K,
                          const float* __restrict__ V,
                          float* __restrict__ ctx,
                          float* __restrict__ att)
{
    // Dynamic LDS: S tile, 16 rows x 2048 cols fp32 = 128 KB (WGP has 320 KB)
    extern __shared__ float S[];
    __shared__ float invS[BQ];   // per-row 1/sum(exp)

    const int tid  = threadIdx.x;
    const int lane = tid & 31;          // wave32
    const int wave = tid >> 5;          // 0..3
    const int m    = lane & 15;         // row (A/M) or col (B/N) index
    const int hi   = lane >> 4;         // half-wave select

    const int b      = blockIdx.x / (LSEQ / BQ);
    const int qt     = blockIdx.x % (LSEQ / BQ);
    const int q_base = qt * BQ;

    const float* Qb = Q + ((size_t)b * LSEQ + q_base) * DDIM;
    const float* Kb = K + (size_t)b * LSEQ * DDIM;
    const float* Vb = V + (size_t)b * LSEQ * DDIM;

    // ---------------- Phase 1: S = Q_tile x K^T ----------------
    // A-matrix 16x4 f32 layout: lane(m,hi): v0=A[m,k0+2hi], v1=A[m,k0+2hi+1]
    // -> contiguous float2 in memory (D fastest dim). Hoist all 16 k-steps.
    v2f aq[16];
#pragma unroll
    for (int kk = 0; kk < 16; ++kk)
        aq[kk] = *(const v2f*)(Qb + m * DDIM + 4 * kk + 2 * hi);

    for (int t = wave; t < LSEQ / 16; t += 4) {     // each wave: every 4th key tile
        v8f acc = {0.f, 0.f, 0.f, 0.f, 0.f, 0.f, 0.f, 0.f};
        // B-matrix 4x16: lane(n=m,hi): v0=B[2hi,n]=Key[t*16+n][k0+2hi], v1=next k
        const float* Kt = Kb + (size_t)(t * 16 + m) * DDIM + 2 * hi;
#pragma unroll
        for (int kk = 0; kk < 16; ++kk) {
            v2f bk = *(const v2f*)(Kt + 4 * kk);
            acc = __builtin_amdgcn_wmma_f32_16x16x4_f32(false, aq[kk], false, bk,
                                                        (short)0, acc, false, false);
        }
        // C/D layout: VGPR r -> row (r + 8*hi), col = lane&15 within the tile
#pragma unroll
        for (int r = 0; r < 8; ++r)
            S[(r + 8 * hi) * LSEQ + t * 16 + m] = acc[r];
    }
    __syncthreads();

    // ---------------- Phase 2: softmax over rows of S ----------------
    // 8 lanes per row (row g = tid>>3); rows are contiguous 8-lane groups in a wave.
    {
        const int g   = tid >> 3;       // 0..15
        const int lin = tid & 7;

        float rmax = -3.402823466e38f;
        for (int it = 0; it < LSEQ / 32; ++it) {
            const int col = it * 32 + lin * 4;
            const float4 sv = *(const float4*)&S[g * LSEQ + col];
            rmax = fmaxf(rmax, fmaxf(fmaxf(sv.x, sv.y), fmaxf(sv.z, sv.w)));
        }
#pragma unroll
        for (int o = 1; o < 8; o <<= 1)
            rmax = fmaxf(rmax, __shfl_xor(rmax, o, 8));

        // exp(S - max): keep UNNORMALIZED exp in LDS; fold 1/sum into phase 3.
        float rsum = 0.f;
        for (int it = 0; it < LSEQ / 32; ++it) {
            const int col = it * 32 + lin * 4;
            float4 sv = *(const float4*)&S[g * LSEQ + col];
            sv.x = __expf(sv.x - rmax);
            sv.y = __expf(sv.y - rmax);
            sv.z = __expf(sv.z - rmax);
            sv.w = __expf(sv.w - rmax);
            *(float4*)&S[g * LSEQ + col] = sv;
            rsum += (sv.x + sv.y) + (sv.z + sv.w);
        }
#pragma unroll
        for (int o = 1; o < 8; o <<= 1)
            rsum += __shfl_xor(rsum, o, 8);
        const float rinv = 1.0f / rsum;
        if (lin == 0) invS[g] = rinv;

        // Stream normalized attention row to global (float4, coalesced 128B/group)
        float* att_row = att + ((size_t)b * LSEQ + q_base + g) * LSEQ;
        for (int it = 0; it < LSEQ / 32; ++it) {
            const int col = it * 32 + lin * 4;
            float4 ev = *(const float4*)&S[g * LSEQ + col];
            ev.x *= rinv; ev.y *= rinv; ev.z *= rinv; ev.w *= rinv;
            *(float4*)(att_row + col) = ev;
        }
    }
    __syncthreads();

    // ---------------- Phase 3: context = P x V ----------------
    // Each wave owns one 16-wide n-tile of D=64; 512 k-steps of WMMA f32 16x16x4.
    {
        const int nt = wave;                        // 0..3
        v8f acc = {0.f, 0.f, 0.f, 0.f, 0.f, 0.f, 0.f, 0.f};
        const float* Sa = &S[m * LSEQ + 2 * hi];    // A from LDS (ds_load_b64)
        const float* Vt = Vb + nt * 16 + m;         // B lane n = m
        for (int kt = 0; kt < LSEQ / 4; ++kt) {
            v2f av = *(const v2f*)(Sa + 4 * kt);
            const int krow = 4 * kt + 2 * hi;
            v2f bv;
            bv[0] = Vt[(size_t)krow * DDIM];
            bv[1] = Vt[(size_t)(krow + 1) * DDIM];
            acc = __builtin_amdgcn_wmma_f32_16x16x4_f32(false, av, false, bv,
                                                        (short)0, acc, false, false);
        }
        float* Cb = ctx + ((size_t)b * LSEQ + q_base) * DDIM + nt * 16 + m;
#pragma unroll
        for (int r = 0; r < 8; ++r)
            Cb[(size_t)(r + 8 * hi) * DDIM] = acc[r] * invS[r + 8 * hi];
    }
}

extern "C" void kernel_launch(void* const* d_in, const int* in_sizes, int n_in,
                              void* d_out, int out_size, void* d_ws, size_t ws_size,
                              hipStream_t stream) {
    (void)in_sizes; (void)n_in; (void)out_size; (void)d_ws; (void)ws_size;
    const float* q = (const float*)d_in[0];
    const float* k = (const float*)d_in[1];
    const float* v = (const float*)d_in[2];
    float* ctx = (float*)d_out;                                   // [B, L, D]
    float* att = ctx + (size_t)NBATCH * LSEQ * DDIM;              // [B, L, L]

    const size_t smem = (size_t)BQ * LSEQ * sizeof(float);        // 128 KB
    hipFuncSetAttribute((const void*)sdpa_wmma_f32_kernel,
                        hipFuncAttributeMaxDynamicSharedMemorySize, (int)smem);

    dim3 grid(NBATCH * (LSEQ / BQ));   // 2048 workgroups
    dim3 block(128);                   // 4 waves (wave32)
    sdpa_wmma_f32_kernel<<<grid, block, smem, stream>>>(q, k, v, ctx, att);
}